// Prompt_3066606649608
// MI455X (gfx1250) — compile-verified
//
#include <hip/hip_runtime.h>
#include <hip/hip_bf16.h>

// ---------------------------------------------------------------------------
// Problem sizes (fixed by the reference): B=128, L=2048, D=768, V=500, P=48, K=4
// ---------------------------------------------------------------------------
#define BB   128
#define LL   2048
#define DD   768
#define VV   500
#define PP   48
#define TOPK 4
#define NCHUNK 16          // L split into 16 chunks of 128 rows for the mean
#define LCH    (LL / NCHUNK)

typedef __attribute__((ext_vector_type(2))) float v2f;
typedef __attribute__((ext_vector_type(4))) float v4f;
typedef __attribute__((ext_vector_type(8))) float v8f;

__device__ __forceinline__ v8f wmma_f32_4(v2f a, v2f b, v8f c) {
    // V_WMMA_F32_16X16X4_F32 : D = A(16x4) * B(4x16) + C(16x16), fp32 throughout
    return __builtin_amdgcn_wmma_f32_16x16x4_f32(
        /*neg_a=*/false, a, /*neg_b=*/false, b,
        /*c_mod=*/(short)0, c, /*reuse_a=*/false, /*reuse_b=*/false);
}

// ---------------------------------------------------------------------------
// 1) Partial sums of x_embed over L (dominant 805 MB stream -> non-temporal)
//    grid (B, NCHUNK) x 192 threads; each thread owns one v4f column of D.
// ---------------------------------------------------------------------------
__global__ void mean_partial_kernel(const float* __restrict__ x,
                                    float* __restrict__ part) {
    const int b = blockIdx.x;
    const int c = blockIdx.y;
    const int t = threadIdx.x;                       // 0..191
    const v4f* src =
        (const v4f*)(x + ((size_t)b * LL + (size_t)c * LCH) * DD);
    v4f acc = {0.f, 0.f, 0.f, 0.f};
#pragma unroll 4
    for (int l = 0; l < LCH; ++l) {
        v4f v = __builtin_nontemporal_load(&src[l * (DD / 4) + t]);
        acc += v;
    }
    v4f* dst = (v4f*)(part + ((size_t)b * NCHUNK + c) * DD);
    dst[t] = acc;
}

// 2) Combine partials -> x_mean.  grid B x 192 threads (v4f per thread).
__global__ void mean_final_kernel(const float* __restrict__ part,
                                  float* __restrict__ xmean) {
    const int b = blockIdx.x;
    const int t = threadIdx.x;
    const v4f* src = (const v4f*)(part + (size_t)b * NCHUNK * DD);
    v4f acc = {0.f, 0.f, 0.f, 0.f};
#pragma unroll
    for (int c = 0; c < NCHUNK; ++c) {
        acc += src[c * (DD / 4) + t];
    }
    const float s = 1.0f / (float)LL;
    v4f* dst = (v4f*)(xmean + (size_t)b * DD);
    dst[t] = acc * s;
}

// ---------------------------------------------------------------------------
// 3) prompt_key[P,D] = W[P,V] @ wte[V,D] + b[:,None]   via FP32 WMMA.
//    144 tiles (3 M-tiles x 48 N-tiles), one wave per tile. 18 blocks x 256.
// ---------------------------------------------------------------------------
__global__ void prompt_key_wmma_kernel(const float* __restrict__ W,
                                       const float* __restrict__ wte,
                                       const float* __restrict__ bias,
                                       float* __restrict__ pk) {
    const int wave = ((blockIdx.x * blockDim.x + threadIdx.x) >> 5);
    const int lane = threadIdx.x & 31;
    const int mt = wave / (DD / 16);   // 0..2
    const int nt = wave % (DD / 16);   // 0..47
    const int m0 = mt * 16, n0 = nt * 16;
    const int half = lane >> 4;        // 0: K={0,1}, 1: K={2,3}
    const int l16  = lane & 15;
    const int kk   = half * 2;

    const float* wrow = W + (size_t)(m0 + l16) * VV;   // A row for this lane
    v8f c = {};
    for (int k = 0; k < VV; k += 4) {                  // 500 = 4*125 exactly
        v2f a = *(const v2f*)(wrow + k + kk);          // 8B aligned (stride 2000B)
        v2f bb;
        bb.x = wte[(size_t)(k + kk)     * DD + n0 + l16];
        bb.y = wte[(size_t)(k + kk + 1) * DD + n0 + l16];
        c = wmma_f32_4(a, bb, c);
    }
    // C/D layout: VGPR i -> row m = i + 8*half, col n = l16
    const int n = n0 + l16;
#pragma unroll
    for (int i = 0; i < 8; ++i) {
        const int m = m0 + i + half * 8;
        pk[(size_t)m * DD + n] = c[i] + bias[m];
    }
}

// 4) Per-prompt inverse L2 norm: rn[p] = rsqrt(max(sum_d pk[p,d]^2, eps))
__global__ void rownorm_kernel(const float* __restrict__ pk,
                               float* __restrict__ rn) {
    __shared__ float sh[256];
    const int p = blockIdx.x;
    const int t = threadIdx.x;
    float acc = 0.f;
#pragma unroll
    for (int i = 0; i < DD / 256; ++i) {
        float v = pk[(size_t)p * DD + t + i * 256];
        acc += v * v;
    }
    sh[t] = acc;
    __syncthreads();
    for (int w = 128; w > 0; w >>= 1) {
        if (t < w) sh[t] += sh[t + w];
        __syncthreads();
    }
    if (t == 0) rn[p] = rsqrtf(fmaxf(sh[0], 1e-12f));
}

// ---------------------------------------------------------------------------
// 5) similarity[B,P] = x_mean[B,D] @ (rn * pk)^T   via FP32 WMMA.
//    24 tiles (8 M-tiles x 3 N-tiles), one wave per tile. 3 blocks x 256.
// ---------------------------------------------------------------------------
__global__ void similarity_wmma_kernel(const float* __restrict__ xmean,
                                       const float* __restrict__ pk,
                                       const float* __restrict__ rn,
                                       float* __restrict__ sim) {
    const int wave = ((blockIdx.x * blockDim.x + threadIdx.x) >> 5);
    const int lane = threadIdx.x & 31;
    const int mt = wave / (PP / 16);   // 0..7
    const int nt = wave % (PP / 16);   // 0..2
    const int m0 = mt * 16, n0 = nt * 16;
    const int half = lane >> 4;
    const int l16  = lane & 15;
    const int kk   = half * 2;

    const float  r    = rn[n0 + l16];
    const float* arow = xmean + (size_t)(m0 + l16) * DD;
    const float* brow = pk    + (size_t)(n0 + l16) * DD;  // B column n = prompt row n
    v8f c = {};
    for (int k = 0; k < DD; k += 4) {                     // 768 = 4*192 exactly
        v2f a  = *(const v2f*)(arow + k + kk);
        v2f pb = *(const v2f*)(brow + k + kk);
        v2f bb; bb.x = pb.x * r; bb.y = pb.y * r;
        c = wmma_f32_4(a, bb, c);
    }
    const int n = n0 + l16;
#pragma unroll
    for (int i = 0; i < 8; ++i) {
        const int m = m0 + i + half * 8;
        sim[(size_t)m * PP + n] = c[i];
    }
}

// 6) Top-4 per batch row (strict '>' => lowest index wins ties, matches lax.top_k)
__global__ void topk_kernel(const float* __restrict__ sim,
                            int* __restrict__ idx) {
    const int b = threadIdx.x;
    if (b >= BB) return;
    const float* row = sim + (size_t)b * PP;
    float v[PP];
#pragma unroll
    for (int i = 0; i < PP; ++i) v[i] = row[i];
    unsigned long long used = 0ull;
#pragma unroll
    for (int k = 0; k < TOPK; ++k) {
        float best = -__builtin_inff();
        int bi = 0;
        for (int i = 0; i < PP; ++i) {
            bool free_i = ((used >> i) & 1ull) == 0ull;
            if (free_i && v[i] > best) { best = v[i]; bi = i; }
        }
        used |= (1ull << bi);
        idx[b * TOPK + k] = bi;
    }
}

// 7) reduce_sim = sum(similarity) / B   -> single scalar
__global__ void reduce_sim_kernel(const float* __restrict__ sim,
                                  float* __restrict__ out_scalar) {
    __shared__ float sh[256];
    const int t = threadIdx.x;
    float acc = 0.f;
    for (int i = t; i < BB * PP; i += 256) acc += sim[i];
    sh[t] = acc;
    __syncthreads();
    for (int w = 128; w > 0; w >>= 1) {
        if (t < w) sh[t] += sh[t + w];
        __syncthreads();
    }
    if (t == 0) out_scalar[0] = sh[0] * (1.0f / (float)BB);
}

// 8) Gather raw prompt_key rows: out[b,k,:] = pk[idx[b,k],:]
__global__ void gather_kernel(const float* __restrict__ pk,
                              const int* __restrict__ idx,
                              float* __restrict__ out) {
    const int bk = blockIdx.x;            // 0..511
    const int p  = idx[bk];
    const v4f* src = (const v4f*)(pk + (size_t)p * DD);
    v4f* dst = (v4f*)(out + (size_t)bk * DD);
    dst[threadIdx.x] = src[threadIdx.x];  // 192 threads * v4f = 768 floats
}

// ---------------------------------------------------------------------------
extern "C" void kernel_launch(void* const* d_in, const int* in_sizes, int n_in,
                              void* d_out, int out_size, void* d_ws, size_t ws_size,
                              hipStream_t stream) {
    const float* x_embed = (const float*)d_in[0];   // [128,2048,768]
    const float* wte     = (const float*)d_in[1];   // [500,768]
    const float* W       = (const float*)d_in[2];   // [48,500]
    const float* bias    = (const float*)d_in[3];   // [48]

    float* out_prompts = (float*)d_out;                       // [128,4,768]
    float* out_scalar  = (float*)d_out + (size_t)BB * TOPK * DD;

    // workspace layout (bytes), all 16B-aligned
    char* ws = (char*)d_ws;
    float* part  = (float*)(ws + 0);                               // 128*16*768*4  = 6,291,456
    float* xmean = (float*)(ws + 6291456);                         // 128*768*4    = 393,216
    float* pk    = (float*)(ws + 6291456 + 393216);                // 48*768*4     = 147,456
    float* rn    = (float*)(ws + 6291456 + 393216 + 147456);       // 48*4 (pad 256)
    float* sim   = (float*)(ws + 6291456 + 393216 + 147456 + 256); // 128*48*4     = 24,576
    int*   idx   = (int*)  (ws + 6291456 + 393216 + 147456 + 256 + 24576); // 512*4

    // (1)-(2) dominant streaming mean over L
    mean_partial_kernel<<<dim3(BB, NCHUNK), 192, 0, stream>>>(x_embed, part);
    mean_final_kernel<<<BB, 192, 0, stream>>>(part, xmean);

    // (3)-(4) prompt_key GEMM (fp32 WMMA) + row norms
    prompt_key_wmma_kernel<<<(3 * (DD / 16)) / 8, 256, 0, stream>>>(W, wte, bias, pk);
    rownorm_kernel<<<PP, 256, 0, stream>>>(pk, rn);

    // (5) similarity GEMM (fp32 WMMA)
    similarity_wmma_kernel<<<((BB / 16) * (PP / 16)) / 8, 256, 0, stream>>>(xmean, pk, rn, sim);

    // (6)-(8) top-k, scalar, gather
    topk_kernel<<<1, BB, 0, stream>>>(sim, idx);
    reduce_sim_kernel<<<1, 256, 0, stream>>>(sim, out_scalar);
    gather_kernel<<<BB * TOPK, DD / 4, 0, stream>>>(pk, idx, out_prompts);
}